// DurationEncoding_2207613190389
// MI455X (gfx1250) — compile-verified
//
#include <hip/hip_runtime.h>
#include <cstdint>
#include <cstddef>

// DurationEncoding: idx = searchsorted(edges, t); out[n,:] = table[idx,:]
// Memory-bound gather (512MB stream out). CDNA5 path: async global->LDS
// staging of the embedding table + edges, LDS-served gathers, NT wide stores.

#define NUM_EDGES 100
#define VOCAB     101
#define DIM       128

typedef float v4f __attribute__((ext_vector_type(4)));

// gfx1250 async copy: global memory -> LDS, 16 bytes per lane, ASYNCcnt-tracked.
__device__ __forceinline__ void async_g2lds_b128(uint32_t lds_byte_off, const void* gptr) {
    asm volatile("global_load_async_to_lds_b128 %0, %1, off"
                 :: "v"(lds_byte_off), "v"(gptr)
                 : "memory");
}

__device__ __forceinline__ void wait_async_all() {
    asm volatile("s_wait_asynccnt 0x0" ::: "memory");
}

__global__ __launch_bounds__(256) void duration_encoding_kernel(
    const float* __restrict__ time_value,   // [n]
    const float* __restrict__ bin_edges,    // [100], sorted
    const float* __restrict__ embed_table,  // [101,128]
    float* __restrict__ out,                // [n,128]
    int n)
{
    __shared__ float sEdges[DIM];           // 100 used, padded to 128 floats
    __shared__ float sTable[VOCAB * DIM];   // 51712 bytes

    const int tid = threadIdx.x;

    // ---- Stage bin_edges into LDS: 25 x 16B async chunks ----
    if (tid < NUM_EDGES / 4) {
        uint32_t l = (uint32_t)(size_t)(&sEdges[tid * 4]);   // low 32 bits = LDS offset
        async_g2lds_b128(l, bin_edges + tid * 4);
    }
    // ---- Stage embed_table into LDS: 3232 x 16B async chunks ----
    for (int c = tid; c < (VOCAB * DIM) / 4; c += (int)blockDim.x) {
        uint32_t l = (uint32_t)(size_t)(&sTable[c * 4]);
        async_g2lds_b128(l, embed_table + c * 4);
    }
    wait_async_all();       // this wave's async copies landed in LDS
    __syncthreads();        // all waves' copies visible

    const int lane          = tid & 31;
    const int waveInBlock   = tid >> 5;
    const int wavesPerBlock = (int)blockDim.x >> 5;
    const int gwave         = (int)blockIdx.x * wavesPerBlock + waveInBlock;
    const int nwaves        = (int)gridDim.x * wavesPerBlock;

    // Each wave owns tiles of 32 rows. Lane L searches row (base+L); then the
    // wave emits the 32 rows, one contiguous 512B row per store instruction.
    for (int base = gwave * 32; base < n; base += nwaves * 32) {
        const int myrow = base + lane;
        float v = (myrow < n) ? time_value[myrow] : 0.0f;   // coalesced 128B

        // lower_bound over LDS edges: first i with edges[i] >= v  (side='left')
        int lo = 0, hi = NUM_EDGES;
        while (lo < hi) {
            int mid = (lo + hi) >> 1;
            if (sEdges[mid] < v) lo = mid + 1; else hi = mid;
        }

        const int limit = (n - base < 32) ? (n - base) : 32;
        for (int r = 0; r < limit; ++r) {
            const int idx = __shfl(lo, r, 32);              // broadcast lane r's bucket
            const v4f d = *(const v4f*)(&sTable[idx * DIM + lane * 4]);  // ds_load_b128
            __builtin_nontemporal_store(                     // NT: zero-reuse stream
                d, (v4f*)(out + (size_t)(base + r) * DIM + lane * 4));
        }
    }
}

extern "C" void kernel_launch(void* const* d_in, const int* in_sizes, int n_in,
                              void* d_out, int out_size, void* d_ws, size_t ws_size,
                              hipStream_t stream) {
    const float* time_value  = (const float*)d_in[0];
    const float* bin_edges   = (const float*)d_in[1];
    const float* embed_table = (const float*)d_in[2];
    float* out = (float*)d_out;
    const int n = in_sizes[0];

    // 2048 blocks x 8 wave32 = 16384 waves; 2 tiles of 32 rows per wave at n=2^20.
    const int threads = 256;
    int blocks = 2048;
    const int rowsPerBlock = (threads / 32) * 32;  // 256 rows per block-tile pass
    int needed = (n + rowsPerBlock - 1) / rowsPerBlock;
    if (blocks > needed) blocks = needed > 0 ? needed : 1;

    duration_encoding_kernel<<<blocks, threads, 0, stream>>>(
        time_value, bin_edges, embed_table, out, n);
}